// HFComp_7103875907593
// MI455X (gfx1250) — compile-verified
//
#include <hip/hip_runtime.h>
#include <math.h>

typedef float v2f __attribute__((ext_vector_type(2)));
typedef float v8f __attribute__((ext_vector_type(8)));

#define HW 65536
#define CDIM 64
#define BATCH 8
#define EPSV 1e-12f

// fp32 WMMA helper: D = A(16x4) * B(4x16) + C, full precision.
__device__ __forceinline__ v8f wmma4(v2f a, v2f b, v8f c) {
    return __builtin_amdgcn_wmma_f32_16x16x4_f32(false, a, false, b, (short)0, c, false, false);
}

// 64x128 = W[64x64] * S[64x128] GEMM per block (8 waves, wave w owns 16 pixels).
// acc layout per wave: acc[mt][r] -> (m = mt*16 + r + 8*lh, n = wv*16 + l16)
__device__ __forceinline__ void gemm64(const float* __restrict__ w,
                                       const float* __restrict__ bias,
                                       const float* s,
                                       int wv, int l16, int lh, v8f acc[4]) {
    for (int mt = 0; mt < 4; ++mt)
        for (int r = 0; r < 8; ++r) acc[mt][r] = bias[mt * 16 + r + 8 * lh];
    const int pl = wv * 16 + l16;
    for (int k0 = 0; k0 < 16; ++k0) {
        const int kk = k0 * 4 + 2 * lh;
        v2f bv; bv.x = s[kk * 128 + pl]; bv.y = s[(kk + 1) * 128 + pl];
        for (int mt = 0; mt < 4; ++mt) {
            const float* wp = w + (mt * 16 + l16) * 64 + kk;
            v2f av; av.x = wp[0]; av.y = wp[1];
            acc[mt] = wmma4(av, bv, acc[mt]);
        }
    }
}

// ---------------- Pass 1: q/k/v 1x1 convs (WMMA) + partial gram / sum-of-squares ----------
// grid (512, 8), block 256. Stores only v; q,k live in LDS only.
__global__ __launch_bounds__(256)
void qkv_kernel(const float* __restrict__ x, const float* __restrict__ y,
                const float* __restrict__ qw, const float* __restrict__ qb,
                const float* __restrict__ kw, const float* __restrict__ kb,
                const float* __restrict__ vw, const float* __restrict__ vbias,
                float* __restrict__ vout, float* __restrict__ part) {
    __shared__ float xs[CDIM * 128];  // 32 KB : x tile, later q tile
    __shared__ float ys[CDIM * 128];  // 32 KB : y tile, later k tile
    const int tid = threadIdx.x;
    const int wv = tid >> 5, ln = tid & 31, l16 = ln & 15, lh = ln >> 4;
    const int b = blockIdx.y;
    const int pixbase = blockIdx.x * 128;

    {   // coalesced b128 tile loads
        const float4* xg = (const float4*)(x + (size_t)b * CDIM * HW);
        const float4* yg = (const float4*)(y + (size_t)b * CDIM * HW);
        float4* xs4 = (float4*)xs; float4* ys4 = (float4*)ys;
        for (int i = tid; i < CDIM * 32; i += 256) {
            const int c = i >> 5, p4 = i & 31;
            const size_t g = (size_t)c * (HW / 4) + (pixbase >> 2) + p4;
            xs4[i] = xg[g];
            ys4[i] = yg[g];
        }
    }
    __syncthreads();

    v8f qa[4];
    gemm64(qw, qb, xs, wv, l16, lh, qa);
    __syncthreads();                       // everyone done reading xs
    for (int mt = 0; mt < 4; ++mt)         // overwrite xs with q tile [m][p]
        for (int r = 0; r < 8; ++r)
            xs[(mt * 16 + r + 8 * lh) * 128 + wv * 16 + l16] = qa[mt][r];

    v8f ka[4], va[4];
    gemm64(kw, kb, ys, wv, l16, lh, ka);
    gemm64(vw, vbias, ys, wv, l16, lh, va);

    {   // store v (only tensor that must reach HBM from this pass)
        float* vo = vout + (size_t)b * CDIM * HW + pixbase + wv * 16 + l16;
        for (int mt = 0; mt < 4; ++mt)
            for (int r = 0; r < 8; ++r)
                vo[(size_t)(mt * 16 + r + 8 * lh) * HW] = va[mt][r];
    }
    __syncthreads();                       // everyone done reading ys
    for (int mt = 0; mt < 4; ++mt)         // overwrite ys with k tile [m][p]
        for (int r = 0; r < 8; ++r)
            ys[(mt * 16 + r + 8 * lh) * 128 + wv * 16 + l16] = ka[mt][r];
    __syncthreads();

    // part layout: part[b][o][512 blocks], o: 0..1023 = G (h*256+c*16+d),
    //              1024..1087 = sum q^2 per channel, 1088..1151 = sum k^2.
    float* pb = part + (size_t)b * 1152 * 512 + blockIdx.x;
    if (wv < 4) {                          // gram partials, one head per wave
        const int h = wv, d = l16;
        const float* krow = ys + (h * 16 + d) * 128;
        float s[8];
        for (int e = 0; e < 8; ++e) s[e] = 0.f;
        for (int p = 0; p < 128; ++p) {
            const float kv = krow[p];
            for (int e = 0; e < 8; ++e)
                s[e] = fmaf(xs[(h * 16 + 2 * e + lh) * 128 + p], kv, s[e]);
        }
        for (int e = 0; e < 8; ++e) {
            const int c = 2 * e + lh;
            pb[(size_t)(h * 256 + c * 16 + d) * 512] = s[e];
        }
    } else if (wv < 6) {                   // q sum-of-squares
        const int c = (wv - 4) * 32 + ln;
        const float* row = xs + c * 128;
        float s = 0.f;
        for (int p = 0; p < 128; ++p) s = fmaf(row[p], row[p], s);
        pb[(size_t)(1024 + c) * 512] = s;
    } else {                               // k sum-of-squares
        const int c = (wv - 6) * 32 + ln;
        const float* row = ys + c * 128;
        float s = 0.f;
        for (int p = 0; p < 128; ++p) s = fmaf(row[p], row[p], s);
        pb[(size_t)(1088 + c) * 512] = s;
    }
}

// ---------------- Pass 2: deterministic partial reduction -----------------------------
__global__ __launch_bounds__(256)
void reduce_kernel(const float* __restrict__ part, float* __restrict__ G,
                   float* __restrict__ qn, float* __restrict__ kn) {
    const int b = blockIdx.x;
    for (int o = threadIdx.x; o < 1152; o += 256) {
        const float* p = part + ((size_t)b * 1152 + o) * 512;
        float s = 0.f;
        for (int i = 0; i < 512; ++i) s += p[i];   // fixed order -> deterministic
        if (o < 1024)      G[b * 1024 + o] = s;
        else if (o < 1088) qn[b * 64 + (o - 1024)] = sqrtf(s);
        else               kn[b * 64 + (o - 1088)] = sqrtf(s);
    }
}

// ---------------- Pass 3: scale by norms + softmax (16-wide rows) ----------------------
__global__ void attn_kernel(const float* __restrict__ G, const float* __restrict__ qn,
                            const float* __restrict__ kn, float* __restrict__ attn) {
    const int bh = blockIdx.x, b = bh >> 2, h = bh & 3;
    const int c = threadIdx.x;
    if (c >= 16) return;
    const float* g = G + b * 1024 + h * 256 + c * 16;
    const float qv = fmaxf(qn[b * 64 + h * 16 + c], EPSV);
    float row[16], mx = -INFINITY;
    for (int d = 0; d < 16; ++d) {
        const float kv = fmaxf(kn[b * 64 + h * 16 + d], EPSV);
        row[d] = g[d] / (qv * kv);
        mx = fmaxf(mx, row[d]);
    }
    float sum = 0.f;
    for (int d = 0; d < 16; ++d) { row[d] = expf(row[d] - mx); sum += row[d]; }
    const float inv = 1.f / sum;
    float* a = attn + bh * 256 + c * 16;
    for (int d = 0; d < 16; ++d) a[d] = row[d] * inv;
}

// ---------------- Pass 4: out = attn @ v (WMMA, streaming) -----------------------------
__global__ __launch_bounds__(256)
void av_kernel(const float* __restrict__ attn, const float* __restrict__ vbuf,
               float* __restrict__ out) {
    const int tid = threadIdx.x;
    const int wv = tid >> 5, ln = tid & 31, l16 = ln & 15, lh = ln >> 4;
    const int b = blockIdx.y;
    const int p = blockIdx.x * 128 + wv * 16 + l16;
    const float* vp = vbuf + (size_t)b * CDIM * HW + p;
    float* op = out + (size_t)b * CDIM * HW + p;
    for (int h = 0; h < 4; ++h) {
        const float* a = attn + (b * 4 + h) * 256;
        v8f acc;
        for (int r = 0; r < 8; ++r) acc[r] = 0.f;
        for (int k0 = 0; k0 < 4; ++k0) {
            const int kk = k0 * 4 + 2 * lh;
            v2f av; av.x = a[l16 * 16 + kk]; av.y = a[l16 * 16 + kk + 1];
            v2f bv; bv.x = vp[(size_t)(h * 16 + kk) * HW];
            bv.y = vp[(size_t)(h * 16 + kk + 1) * HW];
            acc = wmma4(av, bv, acc);
        }
        for (int r = 0; r < 8; ++r)
            op[(size_t)(h * 16 + r + 8 * lh) * HW] = acc[r];
    }
}

// ---------------- Pass 5..8: 3x3 conv as implicit GEMM (dominant FLOPs) ----------------
// tile = 32x4 output pixels, halo 34x6 in LDS (52 KB). K = 9 taps x 64ch.
template <bool RELU>
__global__ __launch_bounds__(256)
void conv3_kernel(const float* __restrict__ in, const float* __restrict__ wgt,
                  const float* __restrict__ bias, const float* __restrict__ res,
                  const float* __restrict__ addy, float* __restrict__ out) {
    __shared__ float ins[CDIM * 6 * 34];  // 52224 B
    const int tid = threadIdx.x;
    const int wv = tid >> 5, ln = tid & 31, l16 = ln & 15, lh = ln >> 4;
    const int b = blockIdx.y;
    const int tx0 = (blockIdx.x & 7) * 32;
    const int ty0 = (blockIdx.x >> 3) * 4;

    // L2-resident weights: 147 KB -> speculative prefetch (global_prefetch_b8)
    for (int i = tid; i < 1152; i += 256)
        __builtin_prefetch((const char*)wgt + (size_t)i * 128, 0, 1);

    const float* inb = in + (size_t)b * CDIM * HW;
    for (int i = tid; i < CDIM * 204; i += 256) {      // halo load w/ zero pad
        const int c = i / 204, rem = i % 204;
        const int yy = rem / 34, xx = rem % 34;
        const int gy = ty0 + yy - 1, gx = tx0 + xx - 1;
        float val = 0.f;
        if (gy >= 0 && gy < 256 && gx >= 0 && gx < 256)
            val = inb[(size_t)c * HW + gy * 256 + gx];
        ins[i] = val;
    }
    __syncthreads();

    const int ox = (wv & 1) * 16 + l16;
    const int oy = wv >> 1;
    v8f acc[4];
    for (int mt = 0; mt < 4; ++mt)
        for (int r = 0; r < 8; ++r) acc[mt][r] = bias[mt * 16 + r + 8 * lh];

    for (int ty = 0; ty < 3; ++ty)
        for (int tx = 0; tx < 3; ++tx) {
            const float* sp = ins + (oy + ty) * 34 + (ox + tx);
            const float* wp0 = wgt + ty * 3 + tx;
            for (int k0 = 0; k0 < 16; ++k0) {
                const int kk = k0 * 4 + 2 * lh;
                v2f bv; bv.x = sp[kk * 204]; bv.y = sp[(kk + 1) * 204];
                for (int mt = 0; mt < 4; ++mt) {
                    const float* wp = wp0 + ((size_t)(mt * 16 + l16) * 64 + kk) * 9;
                    v2f av; av.x = wp[0]; av.y = wp[9];
                    acc[mt] = wmma4(av, bv, acc[mt]);
                }
            }
        }

    const size_t gp0 = (size_t)b * CDIM * HW + (size_t)(ty0 + oy) * 256 + tx0 + ox;
    for (int mt = 0; mt < 4; ++mt)
        for (int r = 0; r < 8; ++r) {
            const int m = mt * 16 + r + 8 * lh;
            float val = acc[mt][r];
            if (RELU) val = fmaxf(val, 0.f);
            const size_t gp = gp0 + (size_t)m * HW;
            if (res) val += res[gp];
            if (addy) val += addy[gp];
            out[gp] = val;
        }
}

extern "C" void kernel_launch(void* const* d_in, const int* in_sizes, int n_in,
                              void* d_out, int out_size, void* d_ws, size_t ws_size,
                              hipStream_t stream) {
    const float* x    = (const float*)d_in[0];
    const float* y    = (const float*)d_in[1];
    const float* qw   = (const float*)d_in[2];
    const float* qb   = (const float*)d_in[3];
    const float* kw   = (const float*)d_in[4];
    const float* kb   = (const float*)d_in[5];
    const float* vw   = (const float*)d_in[6];
    const float* vb   = (const float*)d_in[7];
    const float* r1w1 = (const float*)d_in[8];
    const float* r1b1 = (const float*)d_in[9];
    const float* r1w2 = (const float*)d_in[10];
    const float* r1b2 = (const float*)d_in[11];
    const float* r2w1 = (const float*)d_in[12];
    const float* r2b1 = (const float*)d_in[13];
    const float* r2w2 = (const float*)d_in[14];
    const float* r2b2 = (const float*)d_in[15];
    float* outp = (float*)d_out;

    const size_t FM = (size_t)BATCH * CDIM * HW;  // 33.5M floats per feature map
    char* ws = (char*)d_ws;
    float* bufA = (float*)ws;                 // v, then conv intermediates t1/t2
    float* bufB = (float*)(ws + FM * 4);      // attention output
    float* bufC = (float*)(ws + FM * 8);      // r1
    float* part = (float*)(ws + FM * 12);     // 8*1152*512 partials
    float* Gb   = part + (size_t)8 * 1152 * 512;
    float* qn   = Gb + 8 * 1024;
    float* kn   = qn + 8 * 64;
    float* attn = kn + 8 * 64;

    const dim3 grid(512, 8), blk(256);
    qkv_kernel<<<grid, blk, 0, stream>>>(x, y, qw, qb, kw, kb, vw, vb, bufA, part);
    reduce_kernel<<<dim3(8), blk, 0, stream>>>(part, Gb, qn, kn);
    attn_kernel<<<dim3(32), dim3(32), 0, stream>>>(Gb, qn, kn, attn);
    av_kernel<<<grid, blk, 0, stream>>>(attn, bufA, bufB);
    // ResBlock 1: r1 = out + conv2(relu(conv1(out)))
    conv3_kernel<true ><<<grid, blk, 0, stream>>>(bufB, r1w1, r1b1, nullptr, nullptr, bufA);
    conv3_kernel<false><<<grid, blk, 0, stream>>>(bufA, r1w2, r1b2, bufB, nullptr, bufC);
    // ResBlock 2 + final (+y), fused into the last epilogue
    conv3_kernel<true ><<<grid, blk, 0, stream>>>(bufC, r2w1, r2b1, nullptr, nullptr, bufA);
    conv3_kernel<false><<<grid, blk, 0, stream>>>(bufA, r2w2, r2b2, bufC, y, outp);
}